// RpnTarget_21517786153631
// MI455X (gfx1250) — compile-verified
//
#include <hip/hip_runtime.h>
#include <cstdint>
#include <math.h>

#define BB 8
#define NN 131072
#define MM 64
#define THREADS 256
#define APT 2   // anchors per thread

typedef uint32_t u32;
typedef __attribute__((ext_vector_type(4))) u32 v4u;
typedef __attribute__((ext_vector_type(8))) int v8i;
typedef __attribute__((ext_vector_type(4))) int v4i;

// RPN target assignment: 2 anchors per thread (dual dependency chains for VOPD
// dual-issue), 64 GT boxes staged in LDS via the Tensor Data Mover.
__global__ __launch_bounds__(THREADS) void rpn_targets_kernel(
    const float* __restrict__ anchors,   // (B, N, 4)
    const float* __restrict__ gt,        // (B, M, 5)  [x1,y1,x2,y2,valid]
    float* __restrict__ out)             // reg (B,N,5) flat, then cls (B,N,2) flat
{
    __shared__ __align__(16) float raw[MM * 5];  // packed gt rows, TDM destination
    __shared__ __align__(16) float sb[MM * 8];   // padded: x1,y1,x2,y2,areaOrNaN,0,0,0

    const int b  = blockIdx.y;
    const int n0 = blockIdx.x * (THREADS * APT) + (int)threadIdx.x;   // anchor 0
    const int n1 = n0 + THREADS;                                      // anchor 1
    const float* gtb = gt + (size_t)b * MM * 5;

    // ---- Stage 1: DMA this batch's GT boxes into LDS with the TDM ----
#if __has_builtin(__builtin_amdgcn_tensor_load_to_lds) && __has_builtin(__builtin_amdgcn_s_wait_tensorcnt)
    if (threadIdx.x == 0) {  // wave 0 issues; tensor ops ignore EXEC
        const uint64_t ga  = (uint64_t)(uintptr_t)gtb;        // 57-bit global byte addr
        const u32      lds = (u32)(uintptr_t)&raw[0];         // low 32 bits = LDS offset
        v4u g0;
        g0[0] = 1u;                                           // count=1, user mode
        g0[1] = lds;                                          // lds_addr
        g0[2] = (u32)(ga & 0xFFFFFFFFu);                      // global_addr[31:0]
        g0[3] = (u32)((ga >> 32) & 0x01FFFFFFu) | (2u << 30); // global_addr[56:32], type=2
        v8i g1;
        g1[0] = (int)(2u << 16);                 // workgroup_mask=0, data_size=2 (4 bytes)
        g1[1] = (int)((u32)(MM * 5) << 16);      // tensor_dim0[15:0] = 320
        g1[2] = (int)(1u << 16);                 // tensor_dim0 hi = 0, tensor_dim1 = 1
        g1[3] = (int)((u32)(MM * 5) << 16);      // tensor_dim1 hi = 0, tile_dim0 = 320
        g1[4] = 1;                               // tile_dim1 = 1 (explicit single row)
        g1[5] = MM * 5;                          // tensor_dim0_stride[31:0] = 320
        g1[6] = 0;
        g1[7] = 0;
        v4i gz = {0, 0, 0, 0};                   // <=2D tensor: groups 2/3 unused
#if defined(__clang_major__) && (__clang_major__ >= 23)
        v8i gz8 = {0, 0, 0, 0, 0, 0, 0, 0};
        __builtin_amdgcn_tensor_load_to_lds(g0, g1, gz, gz, gz8, 0);
#else
        __builtin_amdgcn_tensor_load_to_lds(g0, g1, gz, gz, 0);
#endif
        __builtin_amdgcn_s_wait_tensorcnt(0);
    }
#else
    for (int i = threadIdx.x; i < MM * 5; i += THREADS) raw[i] = gtb[i];
#endif
    __syncthreads();

    // ---- Stage 2: repack boxes to 32B stride; fold validity into area as NaN ----
    if (threadIdx.x < MM) {
        const int m = threadIdx.x;
        const float x1 = raw[m * 5 + 0], y1 = raw[m * 5 + 1];
        const float x2 = raw[m * 5 + 2], y2 = raw[m * 5 + 3];
        const float vs = raw[m * 5 + 4];
        const float ab = fmaxf(x2 - x1, 0.f) * fmaxf(y2 - y1, 0.f);
        sb[m * 8 + 0] = x1; sb[m * 8 + 1] = y1;
        sb[m * 8 + 2] = x2; sb[m * 8 + 3] = y2;
        sb[m * 8 + 4] = (vs > 0.f) ? ab : __builtin_nanf(""); // NaN => can never win argmax
        sb[m * 8 + 5] = 0.f; sb[m * 8 + 6] = 0.f; sb[m * 8 + 7] = 0.f;
    }
    __syncthreads();

    // ---- Stage 3: per-anchor IoU argmax over 64 boxes (division-free) ----
    const float4 a0 = *(const float4*)(anchors + ((size_t)b * NN + n0) * 4);
    const float4 a1 = *(const float4*)(anchors + ((size_t)b * NN + n1) * 4);
    const float areaA0 = fmaxf(a0.z - a0.x, 0.f) * fmaxf(a0.w - a0.y, 0.f);
    const float areaA1 = fmaxf(a1.z - a1.x, 0.f) * fmaxf(a1.w - a1.y, 0.f);

    float bI0 = -1.0f, bU0 = 1.0f;  int bM0 = 0;   // sentinel pair == iou -1
    float bI1 = -1.0f, bU1 = 1.0f;  int bM1 = 0;

#pragma unroll 8
    for (int m = 0; m < MM; ++m) {
        const float4 c  = *(const float4*)(sb + m * 8);      // ds_load_b128 broadcast
        const float  ab = sb[m * 8 + 4];                     // areaB or NaN

        // anchor 0 chain
        const float iw0 = fmaxf(fminf(a0.z, c.z) - fmaxf(a0.x, c.x), 0.f);
        const float ih0 = fmaxf(fminf(a0.w, c.w) - fmaxf(a0.y, c.y), 0.f);
        const float in0 = iw0 * ih0;
        const float un0 = (areaA0 + ab) - in0;               // NaN if box invalid
        const bool  g0_ = (in0 * bU0) > (bI0 * un0);         // NaN -> false; '>' keeps first max
        bI0 = g0_ ? in0 : bI0;  bU0 = g0_ ? un0 : bU0;  bM0 = g0_ ? m : bM0;

        // anchor 1 chain (independent -> dual-issue)
        const float iw1 = fmaxf(fminf(a1.z, c.z) - fmaxf(a1.x, c.x), 0.f);
        const float ih1 = fmaxf(fminf(a1.w, c.w) - fmaxf(a1.y, c.y), 0.f);
        const float in1 = iw1 * ih1;
        const float un1 = (areaA1 + ab) - in1;
        const bool  g1_ = (in1 * bU1) > (bI1 * un1);
        bI1 = g1_ ? in1 : bI1;  bU1 = g1_ ? un1 : bU1;  bM1 = g1_ ? m : bM1;
    }

    const size_t base = (size_t)b * NN;
    const size_t clsBase = (size_t)BB * NN * 5;

    // ---- Stage 4: epilogue + stores, per anchor ----
    {
        const float4 mb = *(const float4*)(sb + bM0 * 8);
        const bool pos = bI0 >= 0.5f * bU0;
        const bool ign = bI0 >= 0.3f * bU0;
        const float st = pos ? 1.0f : (ign ? -1.0f : 0.0f);
        const float axc = (a0.x + a0.z) * 0.5f, ayc = (a0.y + a0.w) * 0.5f;
        const float aw  = a0.z - a0.x,          ah  = a0.w - a0.y;
        const float bxc = (mb.x + mb.z) * 0.5f, byc = (mb.y + mb.w) * 0.5f;
        const float bw  = mb.z - mb.x,          bh  = mb.w - mb.y;
        float* ro = out + (base + n0) * 5;
        ro[0] = (bxc - axc) / aw;
        ro[1] = (byc - ayc) / ah;
        ro[2] = logf(bw / aw);
        ro[3] = logf(bh / ah);
        ro[4] = st;
        float2 cv; cv.x = 1.0f; cv.y = st;   // one_hot(0, 1) == 1.0 always
        *(float2*)(out + clsBase + (base + n0) * 2) = cv;
    }
    {
        const float4 mb = *(const float4*)(sb + bM1 * 8);
        const bool pos = bI1 >= 0.5f * bU1;
        const bool ign = bI1 >= 0.3f * bU1;
        const float st = pos ? 1.0f : (ign ? -1.0f : 0.0f);
        const float axc = (a1.x + a1.z) * 0.5f, ayc = (a1.y + a1.w) * 0.5f;
        const float aw  = a1.z - a1.x,          ah  = a1.w - a1.y;
        const float bxc = (mb.x + mb.z) * 0.5f, byc = (mb.y + mb.w) * 0.5f;
        const float bw  = mb.z - mb.x,          bh  = mb.w - mb.y;
        float* ro = out + (base + n1) * 5;
        ro[0] = (bxc - axc) / aw;
        ro[1] = (byc - ayc) / ah;
        ro[2] = logf(bw / aw);
        ro[3] = logf(bh / ah);
        ro[4] = st;
        float2 cv; cv.x = 1.0f; cv.y = st;
        *(float2*)(out + clsBase + (base + n1) * 2) = cv;
    }
}

extern "C" void kernel_launch(void* const* d_in, const int* in_sizes, int n_in,
                              void* d_out, int out_size, void* d_ws, size_t ws_size,
                              hipStream_t stream) {
    (void)in_sizes; (void)n_in; (void)d_ws; (void)ws_size; (void)out_size;
    const float* anchors = (const float*)d_in[0];   // (B,N,4) f32
    const float* gt      = (const float*)d_in[1];   // (B,M,5) f32
    // d_in[2] (gt_class_idxes) is dead: reference zeroes it before one-hot.
    float* out = (float*)d_out;
    dim3 grid(NN / (THREADS * APT), BB);
    rpn_targets_kernel<<<grid, THREADS, 0, stream>>>(anchors, gt, out);
}